// DN4_Layer_27393301414374
// MI455X (gfx1250) — compile-verified
//
#include <hip/hip_runtime.h>
#include <hip/hip_bf16.h>
#include <math.h>

typedef float v2f __attribute__((ext_vector_type(2)));
typedef float v8f __attribute__((ext_vector_type(8)));

#define TDIM 8
#define WQ   75
#define WAY  5
#define SHOT 5
#define CH   640
#define HW   100
#define KTOT 500          // SHOT*HW columns of B
#define PT   112          // padded M (7 tiles of 16)
#define KT   128          // N tile (4 tiles cover 500)
#define KCH  64           // K chunk (of CH=640)
#define SAP  68           // LDS row stride for A  [112 x 68], stride%64==4 -> conflict-free frag loads
#define SBP  68           // LDS row stride for B  [128 x 68]
#define SR   129          // LDS stride for rel    [112 x 129]
#define SMEM_FLOATS (PT * SAP + KT * SBP)     // 16320 floats = 65280 B
static_assert(PT * SR <= SMEM_FLOATS, "relS overlay must fit");
static_assert(15360 + HW <= SMEM_FLOATS, "rowsum overlay must fit");

// ---------------------------------------------------------------------------
// Kernel 1: qInv[tq][c] = 1 / max(||query[tq, c, :]||_2, eps)   (norm over hw)
// ---------------------------------------------------------------------------
__global__ void qnorm_kernel(const float* __restrict__ q, float* __restrict__ qInv) {
    int b = blockIdx.x;                       // t*WQ + q  (600 blocks)
    const float* base = q + (size_t)b * CH * HW;
    for (int c = threadIdx.x; c < CH; c += blockDim.x) {
        const float* p = base + c * HW;
        float ss = 0.0f;
#pragma unroll 4
        for (int i = 0; i < HW; ++i) { float v = p[i]; ss = fmaf(v, v, ss); }
        qInv[(size_t)b * CH + c] = 1.0f / fmaxf(sqrtf(ss), 1e-12f);
    }
}

// ---------------------------------------------------------------------------
// Kernel 2: sInv[t][w][k] = 1 / max(||support[t, w*5+shot, :, hwp]||_2, eps)
// (norm over channels) via LDS float atomics (ds_add_f32).
// ---------------------------------------------------------------------------
__global__ void snorm_kernel(const float* __restrict__ s, float* __restrict__ sInv) {
    __shared__ float ssq[HW];
    int img = blockIdx.x;                     // t*25 + (w*SHOT+shot)  (200 blocks)
    for (int i = threadIdx.x; i < HW; i += blockDim.x) ssq[i] = 0.0f;
    __syncthreads();
    const float* base = s + (size_t)img * CH * HW;
    for (int idx = threadIdx.x; idx < CH * HW; idx += blockDim.x) {
        float v = base[idx];
        atomicAdd(&ssq[idx % HW], v * v);     // ds_add_f32
    }
    __syncthreads();
    if (threadIdx.x < HW) {
        int t = img / (WAY * SHOT), sidx = img % (WAY * SHOT);
        int w = sidx / SHOT, shot = sidx % SHOT;
        int k = shot * HW + threadIdx.x;
        sInv[((size_t)t * WAY + w) * KTOT + k] =
            1.0f / fmaxf(sqrtf(ssq[threadIdx.x]), 1e-12f);
    }
}

// ---------------------------------------------------------------------------
// Kernel 3: per (t,q,way): rel = A(100x640) * B(640x500) via fp32 WMMA,
// running per-row top-3 across N tiles, then sum -> out[(t*75+q)*5+w].
// LDS layouts are row-major with stride 68 so each WMMA fragment is one
// aligned, bank-conflict-free ds_load_b64 straight into an even VGPR pair.
// ---------------------------------------------------------------------------
__global__ void __launch_bounds__(256)
dn4_main_kernel(const float* __restrict__ q, const float* __restrict__ s,
                const float* __restrict__ qInv, const float* __restrict__ sInv,
                float* __restrict__ out) {
    __shared__ __align__(16) float smem[SMEM_FLOATS];
    float* As2    = smem;                     // [PT][SAP]  p-major
    float* Bs2    = smem + PT * SAP;          // [KT][SBP]  k-major
    float* relS   = smem;                     // [PT][SR]   (overlays staging)
    float* rowsum = smem + 15360;             // beyond relS end (14448)

    int bid = blockIdx.x;                     // ((t*WQ+q)*WAY + w), 3000 blocks
    int w   = bid % WAY;
    int tq  = bid / WAY;
    int t   = tq / WQ;

    const float* qbase = q    + (size_t)tq * CH * HW;
    const float* qinv  = qInv + (size_t)tq * CH;
    const float* sinv  = sInv + ((size_t)t * WAY + w) * KTOT;
    const float* sbase = s    + ((size_t)(t * WAY * SHOT + w * SHOT)) * CH * HW;

    int tid    = threadIdx.x;
    int wv     = tid >> 5;
    int lane   = tid & 31;
    int laneN  = lane & 15;
    int laneHi = lane >> 4;                   // 0: K 0..1 / rows 0..7 ; 1: K 2..3 / rows 8..15

    const float* aBase = As2 + (size_t)laneN * SAP + 2 * laneHi;
    const float* bBase = Bs2 + (size_t)(wv * 16 + laneN) * SBP + 2 * laneHi;

    float t0 = -INFINITY, t1 = -INFINITY, t2 = -INFINITY;  // running top-3 (row = tid)

    for (int k0 = 0; k0 < KTOT; k0 += KT) {
        v8f acc[7];
#pragma unroll
        for (int i = 0; i < 7; ++i)
#pragma unroll
            for (int r = 0; r < 8; ++r) acc[i][r] = 0.0f;

        for (int cc = 0; cc < CH; cc += KCH) {
            __syncthreads();                  // prev compute / relS scan done

            // Stage A chunk: As2[p][c] = qn[p][cc+c]; rows >= 100 zero-padded.
            // Global reads contiguous in p (coalesced); LDS stores stride-68.
            for (int idx = tid; idx < KCH * PT; idx += 256) {
                int c = idx / PT, p = idx - c * PT;
                float v = 0.0f;
                if (p < HW) v = qbase[(size_t)(cc + c) * HW + p] * qinv[cc + c];
                As2[p * SAP + c] = v;
            }
            // Stage B chunk: Bs2[k][c] = sn[cc+c][k0+k]; cols >= 500 zero-padded.
            for (int idx = tid; idx < KCH * KT; idx += 256) {
                int c = idx >> 7, k = idx & (KT - 1);
                int gk = k0 + k;
                float v = 0.0f;
                if (gk < KTOT) {
                    int shot = gk / HW, hwp = gk - shot * HW;
                    v = sbase[((size_t)shot * CH + (cc + c)) * HW + hwp] * sinv[gk];
                }
                Bs2[k * SBP + c] = v;
            }
            __syncthreads();

            // 16 K-steps of 4; B fragment (ds_load_b64) reused across 7 M tiles.
#pragma unroll 4
            for (int ck = 0; ck < KCH; ck += 4) {
                v2f b = *(const v2f*)(bBase + ck);
#pragma unroll
                for (int pt = 0; pt < 7; ++pt) {
                    v2f a = *(const v2f*)(aBase + (size_t)pt * 16 * SAP + ck);
                    acc[pt] = __builtin_amdgcn_wmma_f32_16x16x4_f32(
                        false, a, false, b, (short)0, acc[pt], false, false);
                }
            }
        }
        __syncthreads();                      // compute done -> reuse smem as relS

        // Spill D tiles: lane holds column (wv*16+laneN), rows r + 8*laneHi.
#pragma unroll
        for (int pt = 0; pt < 7; ++pt)
#pragma unroll
            for (int r = 0; r < 8; ++r)
                relS[(pt * 16 + laneHi * 8 + r) * SR + wv * 16 + laneN] = acc[pt][r];
        __syncthreads();

        // Running top-3 per row over this N tile (padded cols skipped).
        if (tid < HW) {
            int kmax = (KTOT - k0 < KT) ? (KTOT - k0) : KT;
            const float* row = relS + (size_t)tid * SR;
            for (int k = 0; k < kmax; ++k) {
                float v = row[k];
                if (v > t0)      { t2 = t1; t1 = t0; t0 = v; }
                else if (v > t1) { t2 = t1; t1 = v; }
                else if (v > t2) { t2 = v; }
            }
        }
        __syncthreads();
    }

    if (tid < HW) rowsum[tid] = t0 + t1 + t2;
    __syncthreads();
    if (tid == 0) {
        float ssum = 0.0f;
        for (int i = 0; i < HW; ++i) ssum += rowsum[i];
        out[bid] = ssum;
    }
}

// ---------------------------------------------------------------------------
extern "C" void kernel_launch(void* const* d_in, const int* in_sizes, int n_in,
                              void* d_out, int out_size, void* d_ws, size_t ws_size,
                              hipStream_t stream) {
    const float* qf = (const float*)d_in[0];   // (8, 75, 640, 10, 10)
    const float* sf = (const float*)d_in[1];   // (8, 25, 640, 10, 10)
    float* out  = (float*)d_out;               // (8, 75, 5)
    float* qInv = (float*)d_ws;                                 // 384000 floats
    float* sInv = qInv + (size_t)TDIM * WQ * CH;                // 20000 floats

    qnorm_kernel<<<TDIM * WQ, 128, 0, stream>>>(qf, qInv);
    snorm_kernel<<<TDIM * WAY * SHOT, 256, 0, stream>>>(sf, sInv);
    dn4_main_kernel<<<TDIM * WQ * WAY, 256, 0, stream>>>(qf, sf, qInv, sInv, out);
}